// VQEmbedding_1941325218351
// MI455X (gfx1250) — compile-verified
//
#include <hip/hip_runtime.h>

typedef __attribute__((ext_vector_type(2))) float v2f;
typedef __attribute__((ext_vector_type(8))) float v8f;

namespace {
constexpr int Kc   = 512;
constexpr int Dd   = 256;
constexpr int NSHARED = 10;
constexpr int Bb   = 32;
constexpr int HW   = 64 * 64;
constexpr long DHW = (long)Dd * HW;
constexpr int PIXB = 128;  // pixels per block
constexpr int KT   = 64;   // padded codes per class (real ~50-51)
constexpr int DC   = 64;   // D-chunk staged in LDS
constexpr int PAD  = 66;   // LDS row pitch in floats
}

__global__ __launch_bounds__(256)
void vq_argmin_gather(const float* __restrict__ zin,
                      const float* __restrict__ cbk,
                      const int*   __restrict__ labels,
                      float* __restrict__ out)
{
  __shared__ float Xs[PIXB][PAD];   // x tile, [pixel][d]
  __shared__ float Cs[KT][PAD];     // codebook tile, [code][d]
  __shared__ float cbsq[KT];        // ||c||^2 accumulators
  __shared__ int   idxs[PIXB];      // per-pixel argmin (global code id)

  const int tid  = threadIdx.x;
  const int lane = tid & 31;
  const int wv   = tid >> 5;

  const int blk     = blockIdx.x;
  const int b       = blk >> 5;        // image (32 tiles per image)
  const int tile    = blk & 31;
  const int pixbase = tile * PIXB;
  const int label   = labels[b];

  // Contiguous code range of this class: uniform = round(linspace(-0.5, 9.49, 502)),
  // round-half-even (rintf default mode) matches jnp.round. Shared keys 502..511
  // are always masked in training, so they never enter the range.
  const float step = 9.99f / 501.0f;
  int kstart = 0, kend = 0;
  bool found = false;
  for (int k = 0; k < Kc - NSHARED; ++k) {
    int c = (int)__builtin_rintf(-0.5f + step * (float)k);
    if (c == label) { if (!found) { kstart = k; found = true; } kend = k + 1; }
  }
  const int klen = kend - kstart;      // <= KT

  if (tid < KT) cbsq[tid] = 0.0f;

  v8f acc0 = {}, acc1 = {}, acc2 = {}, acc3 = {};

  const float* zb = zin + (long)b * DHW + pixbase;

  // fp32 WMMA fragment placement (ISA 7.12.2):
  //  A 16x4: lanes 0-15 -> M=0..15 K={0,1}; lanes 16-31 -> same M, K={2,3}
  //  B 4x16: lanes 0-15 -> N=0..15 K={0,1}; lanes 16-31 -> same N, K={2,3}
  const int cn   = lane & 15;
  const int koff = (lane < 16) ? 0 : 2;
  const int prow = wv * 16 + cn;

  for (int dc = 0; dc < Dd; dc += DC) {
    __syncthreads();
    // stage x: global [d][pixel] (contiguous pixels) -> LDS [pixel][d]
    for (int e = tid; e < PIXB * DC; e += 256) {
      int p  = e & (PIXB - 1);
      int dd = e >> 7;
      Xs[p][dd] = zb[(long)(dc + dd) * HW + p];
    }
    // stage codebook tile (row-clamped; pad codes masked via klen later)
    for (int e = tid; e < KT * DC; e += 256) {
      int dd = e & (DC - 1);
      int kk = e >> 6;
      int krow = kstart + kk; if (krow > Kc - 1) krow = Kc - 1;
      Cs[kk][dd] = cbk[(long)krow * Dd + dc + dd];
    }
    __syncthreads();
    // ||c||^2 partial sums from the staged tile (no extra global traffic)
    if (tid < KT) {
      float s = 0.0f;
      #pragma unroll 8
      for (int dd = 0; dd < DC; ++dd) s += Cs[tid][dd] * Cs[tid][dd];
      cbsq[tid] += s;
    }
    // 16(pixels) x 64(codes) fp32 WMMA, K-loop in steps of 4
    #pragma unroll 4
    for (int d4 = 0; d4 < DC; d4 += 4) {
      v2f a, b0, b1, b2, b3;
      a.x  = Xs[prow][d4 + koff];     a.y  = Xs[prow][d4 + koff + 1];
      b0.x = Cs[cn     ][d4 + koff];  b0.y = Cs[cn     ][d4 + koff + 1];
      b1.x = Cs[16 + cn][d4 + koff];  b1.y = Cs[16 + cn][d4 + koff + 1];
      b2.x = Cs[32 + cn][d4 + koff];  b2.y = Cs[32 + cn][d4 + koff + 1];
      b3.x = Cs[48 + cn][d4 + koff];  b3.y = Cs[48 + cn][d4 + koff + 1];
      acc0 = __builtin_amdgcn_wmma_f32_16x16x4_f32(false, a, false, b0, (short)0, acc0, false, false);
      acc1 = __builtin_amdgcn_wmma_f32_16x16x4_f32(false, a, false, b1, (short)0, acc1, false, false);
      acc2 = __builtin_amdgcn_wmma_f32_16x16x4_f32(false, a, false, b2, (short)0, acc2, false, false);
      acc3 = __builtin_amdgcn_wmma_f32_16x16x4_f32(false, a, false, b3, (short)0, acc3, false, false);
    }
  }
  __syncthreads();

  // dist = ||c||^2 - 2 x.c  (||x||^2 is constant per pixel -> dropped for argmin)
  {
    const float INFF = __builtin_inff();
    float cs0 = cbsq[cn], cs1 = cbsq[16 + cn], cs2 = cbsq[32 + cn], cs3 = cbsq[48 + cn];
    #pragma unroll
    for (int j = 0; j < 8; ++j) {
      // C/D layout: VGPR j, lanes 0-15 -> M=j, lanes 16-31 -> M=j+8; N = lane%16
      float d0 = (cn      < klen) ? (cs0 - 2.0f * acc0[j]) : INFF;
      float d1 = (16 + cn < klen) ? (cs1 - 2.0f * acc1[j]) : INFF;
      float d2 = (32 + cn < klen) ? (cs2 - 2.0f * acc2[j]) : INFF;
      float d3 = (48 + cn < klen) ? (cs3 - 2.0f * acc3[j]) : INFF;
      float best = d0; int bi = cn;                 // ascending ids: strict < keeps first
      if (d1 < best) { best = d1; bi = 16 + cn; }
      if (d2 < best) { best = d2; bi = 32 + cn; }
      if (d3 < best) { best = d3; bi = 48 + cn; }
      #pragma unroll
      for (int m = 1; m <= 8; m <<= 1) {           // butterfly within 16-lane halves
        float od = __shfl_xor(best, m, 32);
        int   oi = __shfl_xor(bi,   m, 32);
        if (od < best || (od == best && oi < bi)) { best = od; bi = oi; }
      }
      if (cn == 0) {
        int p = wv * 16 + ((lane >> 4) << 3) + j;
        idxs[p] = kstart + bi;
      }
    }
  }
  __syncthreads();

  // Gather + NHWC->NCHW store. out half 0: x + (c - x) (fp fidelity with the
  // reference's straight-through sum), half 1: c. x re-read hits L2.
  float* outB  = out + (long)b * DHW + pixbase;
  float* outB2 = outB + (long)Bb * DHW;
  for (int e = tid; e < PIXB * Dd; e += 256) {
    int p = e & (PIXB - 1);
    int d = e >> 7;
    float cv = cbk[(long)idxs[p] * Dd + d];
    float xv = zb[(long)d * HW + p];
    long off = (long)d * HW + p;
    outB[off]  = xv + (cv - xv);
    outB2[off] = cv;
  }
}

extern "C" void kernel_launch(void* const* d_in, const int* in_sizes, int n_in,
                              void* d_out, int out_size, void* d_ws, size_t ws_size,
                              hipStream_t stream) {
  const float* z      = (const float*)d_in[0];
  const float* cb     = (const float*)d_in[1];
  const int*   labels = (const int*)d_in[2];
  float*       out    = (float*)d_out;

  dim3 grid(Bb * (HW / PIXB));   // 32 images * 32 tiles = 1024 blocks
  dim3 block(256);               // 8 wave32s
  vq_argmin_gather<<<grid, block, 0, stream>>>(z, cb, labels, out);
}